// Quantizer_189
// MI455X (gfx1250) — compile-verified
//
#include <hip/hip_runtime.h>
#include <float.h>
#include <stdint.h>

// VQ quantizer with D=1: out[i] = nearest codebook value to x[i].
// LUT layout in d_ws (256 floats): [0..127] = midpoint boundaries (bound[0] = -inf),
//                                  [128..255] = sorted codebook values.

#define LDS_AS __attribute__((address_space(3)))

typedef int v4i __attribute__((__vector_size__(16)));

__device__ __forceinline__ void async_load_b128(const void* gsrc, void* lds_dst) {
#if __has_builtin(__builtin_amdgcn_global_load_async_to_lds_b128)
  // Param types per hipcc diagnostic: pointer-to-v4i (16-byte int vector).
  __builtin_amdgcn_global_load_async_to_lds_b128((v4i*)gsrc, (v4i*)lds_dst, 0, 0);
#else
  uint32_t lds_off = (uint32_t)(uintptr_t)(LDS_AS void*)lds_dst;
  uint64_t gaddr   = (uint64_t)(uintptr_t)gsrc;
  asm volatile("global_load_async_to_lds_b128 %0, %1, off"
               :: "v"(lds_off), "v"(gaddr) : "memory");
#endif
}

template <int N>
__device__ __forceinline__ void wait_asynccnt() {
#if __has_builtin(__builtin_amdgcn_s_wait_asynccnt)
  __builtin_amdgcn_s_wait_asynccnt(N);
#else
  asm volatile("s_wait_asynccnt %0" :: "i"(N) : "memory");
#endif
}

// Branchless 7-level binary search over 128 midpoints in LDS; returns nearest code.
__device__ __forceinline__ float nearest_code(float x, const float* lut) {
  int idx = 0;
#pragma unroll
  for (int off = 64; off; off >>= 1) {
    const int p = idx + off;                 // probes indices 1..127 only
    idx = (lut[p] < x) ? p : idx;            // largest idx with bound[idx] < x
  }
  return lut[128 + idx];                     // sorted code value
}

// Main streaming kernel (first in file so the disasm snippet shows it):
// double-buffered async global->LDS x tiles (4KB/tile), 7-probe LDS binary
// search per scalar, float4 stores.
__global__ void __launch_bounds__(256)
vq_main_kernel(const float4* __restrict__ x4, const float* __restrict__ lut_g,
               float4* __restrict__ out4, int tiles) {
  __shared__ float  lut[256];
  __shared__ float4 stage[2][256];
  const int tid = threadIdx.x;
  lut[tid] = lut_g[tid];
  __syncthreads();

  const int stride = (int)gridDim.x;
  int t = (int)blockIdx.x;
  int parity = 0;
  if (t < tiles)
    async_load_b128(&x4[(size_t)t * 256 + tid], &stage[0][tid]);

  for (; t < tiles; t += stride) {
    const int  nt       = t + stride;
    const bool has_next = nt < tiles;
    if (has_next)
      async_load_b128(&x4[(size_t)nt * 256 + tid], &stage[parity ^ 1][tid]);
    if (has_next) wait_asynccnt<1>(); else wait_asynccnt<0>();
    asm volatile("" ::: "memory");         // don't hoist LDS reads above the wait

    const float4 v = stage[parity][tid];   // exactly the bytes this lane staged
    float4 r;
    r.x = nearest_code(v.x, lut);
    r.y = nearest_code(v.y, lut);
    r.z = nearest_code(v.z, lut);
    r.w = nearest_code(v.w, lut);
    out4[(size_t)t * 256 + tid] = r;
    parity ^= 1;
  }
}

// One tiny block: rank-sort 128 codes, emit midpoint boundaries + sorted values.
__global__ void build_lut_kernel(const float* __restrict__ w, float* __restrict__ lut) {
  __shared__ float vals[128];
  __shared__ float sorted[128];
  const int t = threadIdx.x;
  const float v = w[t];
  vals[t] = v;
  __syncthreads();
  int rank = 0;
#pragma unroll 8
  for (int j = 0; j < 128; ++j) {
    const float o = vals[j];
    rank += (int)((o < v) || (o == v && j < t));   // stable total order
  }
  sorted[rank] = v;
  __syncthreads();
  const float b = (rank == 0) ? -FLT_MAX : 0.5f * (sorted[rank - 1] + sorted[rank]);
  lut[rank]       = b;   // boundary between sorted[rank-1] and sorted[rank]
  lut[128 + rank] = v;   // sorted code
}

// Scalar tail (empty for the reference sizes, kept for generality).
__global__ void vq_tail_kernel(const float* __restrict__ x, const float* __restrict__ lut_g,
                               float* __restrict__ out, int n) {
  __shared__ float lut[256];
  if (threadIdx.x < 256) lut[threadIdx.x] = lut_g[threadIdx.x];
  __syncthreads();
  const int i = (int)blockIdx.x * (int)blockDim.x + (int)threadIdx.x;
  if (i < n) out[i] = nearest_code(x[i], lut);
}

extern "C" void kernel_launch(void* const* d_in, const int* in_sizes, int n_in,
                              void* d_out, int out_size, void* d_ws, size_t ws_size,
                              hipStream_t stream) {
  const float* x   = (const float*)d_in[0];   // [16,1,512,512] f32
  const float* w   = (const float*)d_in[1];   // [128,1] f32
  float*       out = (float*)d_out;
  float*       lut = (float*)d_ws;            // 256 floats of scratch

  const int n = in_sizes[0];

  // 1) Sort codebook + midpoints (K is fixed at 128 by the reference).
  build_lut_kernel<<<1, 128, 0, stream>>>(w, lut);

  // 2) Stream quantization: tiles of 256 float4 (1024 floats) per block iteration.
  const int n4    = n / 4;
  const int tiles = n4 / 256;
  if (tiles > 0) {
    const int grid = tiles < 2048 ? tiles : 2048;
    vq_main_kernel<<<grid, 256, 0, stream>>>((const float4*)x, lut, (float4*)out, tiles);
  }

  // 3) Tail (none for 4194304 elements, but stay correct for any shape).
  const int consumed = tiles * 1024;
  const int rem      = n - consumed;
  if (rem > 0) {
    vq_tail_kernel<<<(rem + 255) / 256, 256, 0, stream>>>(x + consumed, lut, out + consumed, rem);
  }
}